// Attn_Pred_Model_47373489275278
// MI455X (gfx1250) — compile-verified
//
#include <hip/hip_runtime.h>

// CDNA5 / gfx1250, wave32. Causal 64-tap decay conv as banded-Toeplitz GEMM on
// V_WMMA_F32_16X16X4_F32 (exact fp32). v4: reversed zero-padded coefficient
// table (A-pairs load in register order, immediate-offset ds_load_2addr, no
// swap movs), strength-reduced loader (one base + immediate offsets; block-0
// zero-fill splits at a compile-time iteration boundary), tile-constant
// mask/bias epilogue (1 fma + 1 store per output).

typedef __attribute__((ext_vector_type(2))) float v2f;
typedef __attribute__((ext_vector_type(8))) float v8f;

#define S_LEN   4096
#define NB_CH   64
#define P_STEPS 64
#define TILE_T  16
#define TPW     4                    // time tiles per wave
#define BLOCK_T (TILE_T * TPW)       // 64 time steps per block
#define WROWS   (P_STEPS + BLOCK_T)  // 128-row x window
#define NCHUNK  (WROWS / 4)          // 32 global K-chunks
#define STR     72                   // padded LDS row stride: 2*72 mod 64 == 16 -> conflict-free half-waves
#define LITERS  ((WROWS * 16) / 128) // 16 loader iterations (8 rows each)
#define ZITERS  (P_STEPS / 8)        // 8: iterations covering t<0 in block 0

__global__ __launch_bounds__(128) void decay_bias_wmma_kernel(
    const float* __restrict__ x,        // (128, 4096, 64)
    const float* __restrict__ pbf,      // (1, 64)
    const float* __restrict__ pbb,      // (1, 64)
    const float* __restrict__ beta_p,   // (1,)
    const float* __restrict__ alpha_p,  // (1,)
    float* __restrict__ out)            // (128, 4096, 64)
{
    __shared__ float xs[WROWS * STR];   // 36.9 KB window [T0-64, T0+64) x 64 ch
    __shared__ float rcoef[128];        // REVERSED zero-padded table: R[p] = coeffpad(78 - p)
    __shared__ float pbf_s[NB_CH];
    __shared__ float pbb_s[NB_CH];

    const int tid = threadIdx.x;
    const int b   = blockIdx.y;             // 0..127
    const int T0  = blockIdx.x * BLOCK_T;   // 0, 64, ..., 4032

    // --- preamble: reversed padded coefficient table + biases into LDS ---
    {
        const float beta  = beta_p[0];
        const float alpha = alpha_p[0];
        const int   j     = 78 - tid;       // coeff exponent; valid in [0,64)
        rcoef[tid] = ((unsigned)j < 64u) ? alpha * __powf(beta, (float)j) : 0.f;
        if (tid < NB_CH) {
            pbf_s[tid] = pbf[tid];
            pbb_s[tid] = pbb[tid];
        }
    }

    // --- cooperative load: fixed bases + compile-time strides (8 rows/iter) ---
    const int row0 = tid >> 4;              // 0..7
    const int c4   = (tid & 15) << 2;       // 0..60
    const float* gsrc = x + (size_t)b * S_LEN * NB_CH
                          + (size_t)(T0 - P_STEPS + row0) * NB_CH + c4;
    float* ldst = &xs[row0 * STR + c4];
    if (blockIdx.x != 0) {                  // common path: no bounds logic at all
        #pragma unroll
        for (int it = 0; it < LITERS; ++it) {
            float4 v = *(const float4*)(gsrc + it * 8 * NB_CH);
            *(float4*)(ldst + it * 8 * STR) = v;
        }
    } else {                                // block 0: t<0 rows == iterations 0..7 exactly
        const float4 z = make_float4(0.f, 0.f, 0.f, 0.f);
        #pragma unroll
        for (int it = 0; it < ZITERS; ++it) {
            *(float4*)(ldst + it * 8 * STR) = z;
        }
        #pragma unroll
        for (int it = ZITERS; it < LITERS; ++it) {
            float4 v = *(const float4*)(gsrc + it * 8 * NB_CH);
            *(float4*)(ldst + it * 8 * STR) = v;
        }
    }
    __syncthreads();

    const int lane = tid & 31;
    const int wave = tid >> 5;          // 4 waves -> channel slices 0,16,32,48
    const int c0   = wave * 16;
    const int m    = lane & 15;         // A row (M) == B col (N) for this lane
    const int kq   = (lane >> 4) << 1;  // lanes 0-15 -> K={0,1}; lanes 16-31 -> K={2,3}

    // --- A-fragments: ascending adjacent pairs from the reversed table ---
    // afr[kk] = { coeffpad(63+m-4kk-kq), coeffpad(62+m-4kk-kq) }
    const int rbase = 15 - m + kq;      // per-lane base; per-kk offsets are immediates
    v2f afr[20];
    #pragma unroll
    for (int kk = 0; kk < 20; ++kk) {
        afr[kk].x = rcoef[rbase + 4 * kk];
        afr[kk].y = rcoef[rbase + 4 * kk + 1];
    }

    // --- main loop over 32 global K-chunks; each B row-pair feeds <=4 WMMAs ---
    v8f acc[TPW];
    #pragma unroll
    for (int j = 0; j < TPW; ++j) acc[j] = (v8f){};

    #pragma unroll
    for (int g = 0; g < NCHUNK; ++g) {
        const int k0 = 4 * g + kq;
        v2f bf;
        bf.x = xs[k0 * STR + c0 + m];
        bf.y = xs[(k0 + 1) * STR + c0 + m];
        #pragma unroll
        for (int j = 0; j < TPW; ++j) {
            const int kk = g - 4 * j;     // compile-time after unroll
            if (kk >= 0 && kk < 20) {
                acc[j] = __builtin_amdgcn_wmma_f32_16x16x4_f32(
                    /*neg_a=*/false, afr[kk], /*neg_b=*/false, bf,
                    /*c_mod=*/(short)0, acc[j], /*reuse_a=*/false, /*reuse_b=*/false);
            }
        }
    }

    // --- epilogue: mask and arange2 are constant within each 16-row tile ---
    // C/D layout: VGPR r -> M = r + 8*(lane>=16), N = lane&15
    const int   thalf = (lane >> 4) << 3;       // 0 or 8
    const int   ch    = c0 + m;
    const float pf    = pbf_s[ch];
    const int   jj    = ch << 6;                // 64 * ch
    float* outp = out + (size_t)b * S_LEN * NB_CH;
    #pragma unroll
    for (int j = 0; j < TPW; ++j) {
        const int   Tj  = T0 + 16 * j;
        const float mk  = (Tj >= jj) ? 1.0f : 0.0f;               // tile-constant mask
        const int   idx = ((Tj - jj + S_LEN) & (S_LEN - 1)) >> 6; // tile-constant arange2
        const float bmk = (pf + pbb_s[idx]) * mk;
        float* po = outp + (size_t)(Tj + thalf) * NB_CH + ch;
        #pragma unroll
        for (int r = 0; r < 8; ++r) {
            po[(size_t)r * NB_CH] = fmaf(acc[j][r], mk, bmk);     // == (acc+bias)*mask
        }
    }
}

extern "C" void kernel_launch(void* const* d_in, const int* in_sizes, int n_in,
                              void* d_out, int out_size, void* d_ws, size_t ws_size,
                              hipStream_t stream) {
    // Input order (setup_inputs): x, pos_bias_fwd, pos_bias_bwd, beta, alpha,
    //                             arange2, mask, past_steps
    const float* x     = (const float*)d_in[0];
    const float* pbf   = (const float*)d_in[1];
    const float* pbb   = (const float*)d_in[2];
    const float* beta  = (const float*)d_in[3];
    const float* alpha = (const float*)d_in[4];
    // arange2 (d_in[5]) and mask (d_in[6]) are pure functions of (t, ch),
    // recomputed on-chip; past_steps (d_in[7]) is fixed at 64.
    float* out = (float*)d_out;

    dim3 grid(S_LEN / BLOCK_T, 128);  // 64 time blocks x 128 batches
    dim3 block(128);                  // 4 wave32s, one 16-channel slice each
    decay_bias_wmma_kernel<<<grid, block, 0, stream>>>(x, pbf, pbb, beta, alpha, out);
}